// EstimatorQNNHybrid_65481071407785
// MI455X (gfx1250) — compile-verified
//
#include <hip/hip_runtime.h>
#include <math.h>

typedef __attribute__((ext_vector_type(2))) float v2f;
typedef __attribute__((ext_vector_type(8))) float v8f;

#define B_TOTAL 8192
#define NSTEP   196

// workspace layout (floats)
#define WS_M    0     // 16x16 circuit matrix (256)
#define WS_PART 256   // per-block partial hx sums: 256 blocks * 4 (1024)
#define WS_CW   1280  // prod(cos(mean_w)) scalar

// ---------------------------------------------------------------------------
// Kernel 1: build the 16x16 real matrix M of the quanv circuit from rl_params.
// Thread j simulates basis column e_j through: for w in 0..3: RY(rl[w]) on
// wire w (wire 0 = MSB of state index), then CNOT(w, (w+1)%4).
// ---------------------------------------------------------------------------
__global__ void build_M(const float* __restrict__ rl, float* __restrict__ ws) {
  int j = threadIdx.x;
  if (j >= 16) return;
  float v[16];
#pragma unroll
  for (int i = 0; i < 16; ++i) v[i] = (i == j) ? 1.f : 0.f;
#pragma unroll
  for (int w = 0; w < 4; ++w) {
    float th = 0.5f * rl[w];
    float ct = __cosf(th), sn = __sinf(th);
    int p = 3 - w;  // bit position of wire w
#pragma unroll
    for (int i0 = 0; i0 < 16; ++i0) {
      if (((i0 >> p) & 1) == 0) {
        int i1 = i0 | (1 << p);
        float a = v[i0], b = v[i1];
        v[i0] = ct * a - sn * b;  // U2 = [[ct,-sn],[sn,ct]]
        v[i1] = sn * a + ct * b;
      }
    }
    int pc = 3 - w, pt = 3 - ((w + 1) & 3);  // CNOT(w, w+1): swap target pair when control set
#pragma unroll
    for (int i = 0; i < 16; ++i) {
      if (((i >> pc) & 1) == 1 && ((i >> pt) & 1) == 0) {
        int i2 = i | (1 << pt);
        float tmp = v[i]; v[i] = v[i2]; v[i2] = tmp;
      }
    }
  }
#pragma unroll
  for (int r = 0; r < 16; ++r) ws[WS_M + r * 16 + j] = v[r];  // row-major M
}

// ---------------------------------------------------------------------------
// Kernel 2: fused quanv (WMMA f32 16x16x4) + 196-step LSTM, one wave32 per
// 32 batch rows. Per step each wave builds two 16-patch tiles S0, computes
// M*S0 with 4 accumulating V_WMMA_F32_16X16X4_F32, squares, contracts with
// ZSIGN (half-row partials combined via shfl_xor 16), then runs the analytic
// qlayer LSTM step per lane. Ends with an in-wave reduction of hx.
// ---------------------------------------------------------------------------
__global__ void __launch_bounds__(32) fused_lstm(
    const float* __restrict__ x, const float* __restrict__ linW,
    const float* __restrict__ linB, const float* __restrict__ qth,
    float* __restrict__ ws) {
  __shared__ float sW[128];
  __shared__ float sB[16];
  __shared__ float sQ[16];
  int lane = threadIdx.x;
  for (int i = lane; i < 128; i += 32) sW[i] = linW[i];
  if (lane < 16) { sB[lane] = linB[lane]; sQ[lane] = qth[lane]; }
  __syncthreads();

  int l = lane & 15;   // tile row (A) / tile column = patch (B,D)
  int h = lane >> 4;   // lane half selects K (A/B) or row group (D)

  // A fragments: M[:,4k:4k+4], ISA layout: lanes0-15 hold K={4k,4k+1}, lanes16-31 K={4k+2,4k+3}
  const float* Mbuf = ws + WS_M;
  v2f a[4];
#pragma unroll
  for (int k = 0; k < 4; ++k) {
    a[k].x = Mbuf[l * 16 + 4 * k + 2 * h];
    a[k].y = Mbuf[l * 16 + 4 * k + 2 * h + 1];
  }

  int bbase = blockIdx.x * 32;
  float hx[4] = {0.f, 0.f, 0.f, 0.f}, cx[4] = {0.f, 0.f, 0.f, 0.f};

  for (int t = 0; t < NSTEP; ++t) {
    int ti = t / 14, tj = t - ti * 14;
    int poff = ti * 56 + tj * 2;  // patch base offset inside a 784-float row
    float qf[2][4];
#pragma unroll
    for (int tile = 0; tile < 2; ++tile) {
      int b = bbase + tile * 16 + l;
      const float* xb = x + (size_t)b * 784 + poff;
      float a0 = xb[0], a1 = xb[1], a2 = xb[28], a3 = xb[29];
      float c0 = __cosf(0.5f * a0), s0 = __sinf(0.5f * a0);
      float c1 = __cosf(0.5f * a1), s1 = __sinf(0.5f * a1);
      float c2 = __cosf(0.5f * a2), s2 = __sinf(0.5f * a2);
      float c3 = __cosf(0.5f * a3), s3 = __sinf(0.5f * a3);
      float p01[4] = {c0 * c1, c0 * s1, s0 * c1, s0 * s1};
      float p23[4] = {c2 * c3, c2 * s3, s2 * c3, s2 * s3};
      float sv[16];
#pragma unroll
      for (int u = 0; u < 4; ++u)
#pragma unroll
        for (int vv = 0; vv < 4; ++vv) sv[u * 4 + vv] = p01[u] * p23[vv];

      v8f acc = {0.f, 0.f, 0.f, 0.f, 0.f, 0.f, 0.f, 0.f};
#pragma unroll
      for (int k = 0; k < 4; ++k) {  // D += M[:,4k:4k+4] * S0[4k:4k+4,:]
        v2f bf;
        bf.x = h ? sv[4 * k + 2] : sv[4 * k + 0];
        bf.y = h ? sv[4 * k + 3] : sv[4 * k + 1];
        acc = __builtin_amdgcn_wmma_f32_16x16x4_f32(
            false, a[k], false, bf, (short)0, acc, false, false);
      }
      // probs -> ZSIGN contraction; this lane holds state rows {0..7}+8h of patch l
      float part[4] = {0.f, 0.f, 0.f, 0.f};
#pragma unroll
      for (int vv = 0; vv < 8; ++vv) {
        float pr = acc[vv] * acc[vv];
        int bidx = vv + 8 * h;
#pragma unroll
        for (int w = 0; w < 4; ++w) {
          float zs = ((bidx >> (3 - w)) & 1) ? -1.f : 1.f;
          part[w] = fmaf(zs, pr, part[w]);
        }
      }
#pragma unroll
      for (int w = 0; w < 4; ++w)
        qf[tile][w] = part[w] + __shfl_xor(part[w], 16, 32);
    }
    // lane owns batch row bbase+lane: tile0 for lanes 0-15, tile1 for 16-31
    float xt[4];
#pragma unroll
    for (int w = 0; w < 4; ++w) xt[w] = (lane < 16) ? qf[0][w] : qf[1][w];

    // LSTM step: gates k = f,i,g,o; qlayer is analytic (CNOT ring = GF(2) map)
    float comb[8] = {xt[0], xt[1], xt[2], xt[3], hx[0], hx[1], hx[2], hx[3]};
    float gate[4][4];
#pragma unroll
    for (int k = 0; k < 4; ++k) {
      float m[4];
#pragma unroll
      for (int w = 0; w < 4; ++w) {
        float ang = sB[k * 4 + w];
#pragma unroll
        for (int j = 0; j < 8; ++j)
          ang = fmaf(comb[j], sW[k * 32 + j * 4 + w], ang);
        m[w] = __cosf(ang + sQ[k * 4 + w]);
      }
      gate[k][0] = m[1] * m[2] * m[3];
      gate[k][1] = m[0] * m[1];
      gate[k][2] = m[0] * m[1] * m[2];
      gate[k][3] = m[0] * m[1] * m[2] * m[3];
    }
#pragma unroll
    for (int w = 0; w < 4; ++w) {
      float f  = 1.f / (1.f + __expf(-gate[0][w]));
      float iv = 1.f / (1.f + __expf(-gate[1][w]));
      float g  = tanhf(gate[2][w]);
      float o  = 1.f / (1.f + __expf(-gate[3][w]));
      cx[w] = f * cx[w] + iv * g;
      hx[w] = o * tanhf(cx[w]);
    }
  }
  // wave reduction of final hx -> per-block partial sums
#pragma unroll
  for (int w = 0; w < 4; ++w) {
    float s = hx[w];
#pragma unroll
    for (int off = 16; off >= 1; off >>= 1) s += __shfl_xor(s, off, 32);
    if (lane == 0) ws[WS_PART + blockIdx.x * 4 + w] = s;
  }
}

// ---------------------------------------------------------------------------
// Kernel 3: deterministic reduction of partials -> prod(cos(mean_w))
// ---------------------------------------------------------------------------
__global__ void finalize(float* __restrict__ ws) {
  __shared__ float sm[4];
  int t = threadIdx.x;
  if (t < 4) {
    float s = 0.f;
    for (int i = 0; i < 256; ++i) s += ws[WS_PART + i * 4 + t];
    sm[t] = __cosf(s * (1.0f / 8192.0f));
  }
  __syncthreads();
  if (t == 0) ws[WS_CW] = sm[0] * sm[1] * sm[2] * sm[3];
}

// ---------------------------------------------------------------------------
// Kernel 4: out[b] = prod_w cos(x[b,w]) * prod_w cos(mean_w)
// ---------------------------------------------------------------------------
__global__ void out_kernel(const float* __restrict__ x,
                           const float* __restrict__ ws,
                           float* __restrict__ out) {
  int b = blockIdx.x * blockDim.x + threadIdx.x;
  if (b >= B_TOTAL) return;
  float cw = ws[WS_CW];
  const float4 xv = *(const float4*)(x + (size_t)b * 784);
  out[b] = cw * __cosf(xv.x) * __cosf(xv.y) * __cosf(xv.z) * __cosf(xv.w);
}

extern "C" void kernel_launch(void* const* d_in, const int* in_sizes, int n_in,
                              void* d_out, int out_size, void* d_ws,
                              size_t ws_size, hipStream_t stream) {
  (void)in_sizes; (void)n_in; (void)out_size; (void)ws_size;
  const float* x  = (const float*)d_in[0];
  const float* rl = (const float*)d_in[1];
  const float* lw = (const float*)d_in[2];
  const float* lb = (const float*)d_in[3];
  const float* qt = (const float*)d_in[4];
  float* ws  = (float*)d_ws;
  float* out = (float*)d_out;

  build_M<<<1, 16, 0, stream>>>(rl, ws);
  fused_lstm<<<256, 32, 0, stream>>>(x, lw, lb, qt, ws);
  finalize<<<1, 32, 0, stream>>>(ws);
  out_kernel<<<32, 256, 0, stream>>>(x, ws, out);
}